// EctLayer_3427383902399
// MI455X (gfx1250) — compile-verified
//
#include <hip/hip_runtime.h>
#include <math.h>

// Problem shape (from reference): N=30000 nodes, E=90000 edges, F=60000 faces,
// B=8 graphs, T=32 thetas, BUMP=32 steps. N/E/F taken dynamically from in_sizes.
#define NB    8
#define NT    32
#define NBUMP 32
#define ACC_ELEMS (NB * NBUMP * NT)   // 8192

typedef __attribute__((ext_vector_type(2))) float v2f;
typedef __attribute__((ext_vector_type(8))) float v8f;

#if __has_builtin(__builtin_amdgcn_exp2f)
#define FAST_EXP2(x) __builtin_amdgcn_exp2f(x)
#else
#define FAST_EXP2(x) exp2f(x)
#endif
#if __has_builtin(__builtin_amdgcn_rcpf)
#define FAST_RCP(x) __builtin_amdgcn_rcpf(x)
#else
#define FAST_RCP(x) (1.0f / (x))
#endif

// ---------------------------------------------------------------------------
// K0: zero the global accumulator in workspace
// ---------------------------------------------------------------------------
__global__ void zero_acc_kernel(float* __restrict__ acc, int n) {
    int i = blockIdx.x * blockDim.x + threadIdx.x;
    if (i < n) acc[i] = 0.0f;
}

// ---------------------------------------------------------------------------
// K1: nh = x @ v   via V_WMMA_F32_16X16X4_F32  (A: 16 nodes x 4 (xyz0),
//     B: 4 x 16 thetas, two theta halves for T=32). One wave per 16-node tile.
//     nh is padded to ntiles*16 rows -> unconditional stores, no exec juggling.
//     Loads use clamped indices + cndmask selects instead of branches.
// ---------------------------------------------------------------------------
__global__ __launch_bounds__(256) void heights_wmma_kernel(
    const float* __restrict__ x, const float* __restrict__ v,
    float* __restrict__ nh, int n_nodes) {
    int lane = threadIdx.x & 31;
    int tile = (blockIdx.x * blockDim.x + threadIdx.x) >> 5;
    int ntiles = (n_nodes + 15) >> 4;
    if (tile >= ntiles) return;          // wave-uniform exit

    bool lo   = (lane < 16);
    int  row  = lane & 15;
    int  node = tile * 16 + row;
    int  nc   = (node < n_nodes) ? node : (n_nodes - 1);   // clamp (pad rows unused)

    // Unconditional loads, then lane-half select (v_cndmask, no exec branches).
    float x0 = x[(size_t)nc * 3 + 0];
    float x1 = x[(size_t)nc * 3 + 1];
    float x2 = x[(size_t)nc * 3 + 2];
    // A: 16x4 f32. lanes 0-15: VGPR0=K0(x), VGPR1=K1(y); lanes 16-31: K2(z), K3(0)
    v2f a;
    a.x = lo ? x0 : x2;
    a.y = lo ? x1 : 0.0f;

    #pragma unroll
    for (int th = 0; th < 2; ++th) {
        int t = th * 16 + row;
        float v0 = v[0 * NT + t];
        float v1 = v[1 * NT + t];
        float v2 = v[2 * NT + t];
        // B: 4x16 f32, lane = theta column; lanes 0-15: K0,K1 rows; 16-31: K2,K3
        v2f b;
        b.x = lo ? v0 : v2;
        b.y = lo ? v1 : 0.0f;

        v8f c = {};
        c = __builtin_amdgcn_wmma_f32_16x16x4_f32(
                /*neg_a=*/false, a, /*neg_b=*/false, b,
                /*c_mod=*/(short)0, c, /*reuse_a=*/false, /*reuse_b=*/false);

        // D: VGPR r -> (M = r or r+8, N = lane&15). Stores unconditional into
        // the padded nh (ntiles*16 rows).
        float* dst = nh + ((size_t)tile * 16 + (lo ? 0 : 8)) * NT + th * 16 + row;
        #pragma unroll
        for (int r = 0; r < 8; ++r) {
            dst[(size_t)r * NT] = c[r];
        }
    }
}

// ---------------------------------------------------------------------------
// K2: ECC accumulation. One wave per simplex (lane = theta). 32 sigmoids per
//     lane per simplex via incremental exp2 argument. LDS [8][32][32] partial
//     sums (ds_add_f32, conflict-free: addr stride = lane), one global
//     f32-atomic flush per block.
// ---------------------------------------------------------------------------
__global__ __launch_bounds__(256) void ecc_accum_kernel(
    const float* __restrict__ nh,
    const int* __restrict__ edge_index,   // [2, E] row-major
    const int* __restrict__ face,         // [3, F] row-major
    const int* __restrict__ batch,        // [N]
    const float* __restrict__ lin,        // [BUMP]
    const int* __restrict__ scale_p,      // [1]
    float* __restrict__ acc,              // [8*32*32] global accumulator
    int n_nodes, int n_edges, int n_faces) {
    __shared__ float lacc[ACC_ELEMS];     // 32 KB

    int tid = threadIdx.x;
    for (int j = tid; j < ACC_ELEMS; j += blockDim.x) lacc[j] = 0.0f;
    __syncthreads();

    const float scale = (float)scale_p[0];
    const float LOG2E = 1.44269504088896340736f;
    const float k     = scale * LOG2E;
    const float lin0  = lin[0];
    const float step  = -k * (lin[1] - lin[0]);   // d/db of exp2 argument

    int lane   = tid & 31;
    int wave   = (blockIdx.x * blockDim.x + tid) >> 5;
    int nwaves = (gridDim.x * blockDim.x) >> 5;
    const int M = n_nodes + n_edges + n_faces;

    for (int m = wave; m < M; m += nwaves) {
        float h, sign;
        int g;
        if (m < n_nodes) {                              // node: +ecc(nh, batch)
            g    = batch[m];
            h    = nh[(size_t)m * NT + lane];
            sign = 1.0f;
        } else if (m < n_nodes + n_edges) {             // edge: -ecc(max2, ...)
            int e  = m - n_nodes;
            int i0 = edge_index[e];
            int i1 = edge_index[n_edges + e];
            g    = batch[i0];
            h    = fmaxf(nh[(size_t)i0 * NT + lane], nh[(size_t)i1 * NT + lane]);
            sign = -1.0f;
        } else {                                        // face: +ecc(max3, ...)
            int f  = m - n_nodes - n_edges;
            int i0 = face[f];
            int i1 = face[n_faces + f];
            int i2 = face[2 * n_faces + f];
            g    = batch[i0];
            h    = fmaxf(fmaxf(nh[(size_t)i0 * NT + lane],
                               nh[(size_t)i1 * NT + lane]),
                         nh[(size_t)i2 * NT + lane]);
            sign = 1.0f;
        }

        // sigmoid(scale*(lin_b - h)) = 1 / (1 + exp2(k*(h - lin_b)))
        float w    = k * (h - lin0);
        int   base = g * (NBUMP * NT) + lane;
        #pragma unroll
        for (int b = 0; b < NBUMP; ++b) {
            float e2  = FAST_EXP2(w);
            float sig = FAST_RCP(1.0f + e2);
            atomicAdd(&lacc[base + b * NT], sign * sig);   // ds_add_f32
            w += step;
        }
    }

    __syncthreads();
    for (int j = tid; j < ACC_ELEMS; j += blockDim.x) {
        unsafeAtomicAdd(&acc[j], lacc[j]);                 // global_atomic_add_f32
    }
}

// ---------------------------------------------------------------------------
// K3: per-graph normalization: out = acc / max(acc over bump,theta)
// ---------------------------------------------------------------------------
__global__ __launch_bounds__(256) void normalize_kernel(
    const float* __restrict__ acc, float* __restrict__ out) {
    __shared__ float red[256];
    int g   = blockIdx.x;
    int tid = threadIdx.x;
    const float* a = acc + (size_t)g * (NBUMP * NT);

    float mx = -INFINITY;
    for (int j = tid; j < NBUMP * NT; j += 256) mx = fmaxf(mx, a[j]);
    red[tid] = mx;
    __syncthreads();
    #pragma unroll
    for (int s = 128; s > 0; s >>= 1) {
        if (tid < s) red[tid] = fmaxf(red[tid], red[tid + s]);
        __syncthreads();
    }
    float m = red[0];
    for (int j = tid; j < NBUMP * NT; j += 256)
        out[(size_t)g * (NBUMP * NT) + j] = a[j] / m;
}

// ---------------------------------------------------------------------------
extern "C" void kernel_launch(void* const* d_in, const int* in_sizes, int n_in,
                              void* d_out, int out_size, void* d_ws, size_t ws_size,
                              hipStream_t stream) {
    const float* x          = (const float*)d_in[0];
    const float* v          = (const float*)d_in[1];
    const float* lin        = (const float*)d_in[2];
    const int*   edge_index = (const int*)d_in[3];
    const int*   face       = (const int*)d_in[4];
    const int*   batch      = (const int*)d_in[5];
    const int*   scale      = (const int*)d_in[6];
    float*       out        = (float*)d_out;

    const int n_nodes = in_sizes[0] / 3;
    const int n_edges = in_sizes[3] / 2;
    const int n_faces = in_sizes[4] / 3;

    // Workspace layout: nh padded to ntiles*16 rows of NT floats, then acc.
    const int ntiles = (n_nodes + 15) / 16;
    float* nh  = (float*)d_ws;
    float* acc = nh + (size_t)ntiles * 16 * NT;

    zero_acc_kernel<<<(ACC_ELEMS + 255) / 256, 256, 0, stream>>>(acc, ACC_ELEMS);

    int blocks1 = (ntiles + 7) / 8;                 // 8 waves (tiles) per block
    heights_wmma_kernel<<<blocks1, 256, 0, stream>>>(x, v, nh, n_nodes);

    ecc_accum_kernel<<<256, 256, 0, stream>>>(nh, edge_index, face, batch, lin,
                                              scale, acc, n_nodes, n_edges, n_faces);

    normalize_kernel<<<NB, 256, 0, stream>>>(acc, out);
}